// QuantumByzantineDetector_16106127360206
// MI455X (gfx1250) — compile-verified
//
#include <hip/hip_runtime.h>
#include <hip/hip_bf16.h>
#include <math.h>

#define NQ      6
#define NLAYERS 5
#define BATCH   1024
#define DIMEN   100000

typedef __attribute__((ext_vector_type(2))) float v2f;
typedef __attribute__((ext_vector_type(8))) float v8f;

// ---------------------------------------------------------------------------
// Kernel A: streaming moment reduction (memory bound: 400MB @ 23.3TB/s ~17us)
// One block per row; f32 per-thread partials, f64 block reduction for Σx³.
// ---------------------------------------------------------------------------
__global__ __launch_bounds__(256) void feat_kernel(const float* __restrict__ x,
                                                   float* __restrict__ angles) {
  const int row = blockIdx.x;
  const int tid = threadIdx.x;
  const float4* __restrict__ xr = (const float4*)(x + (size_t)row * DIMEN);
  const int n4 = DIMEN / 4;

  float s1 = 0.f, s2 = 0.f, s3 = 0.f, ma = 0.f;
  for (int i = tid; i < n4; i += 256) {
    float4 v = xr[i];
    // nan_to_num(x): NaN/±inf -> 0 (comparison is false for NaN)
    float a = (fabsf(v.x) <= 3.402823466e38f) ? v.x : 0.f;
    float b = (fabsf(v.y) <= 3.402823466e38f) ? v.y : 0.f;
    float c = (fabsf(v.z) <= 3.402823466e38f) ? v.z : 0.f;
    float d = (fabsf(v.w) <= 3.402823466e38f) ? v.w : 0.f;
    s1 += a + b + c + d;
    s2 = fmaf(a, a, s2); s2 = fmaf(b, b, s2); s2 = fmaf(c, c, s2); s2 = fmaf(d, d, s2);
    s3 = fmaf(a * a, a, s3); s3 = fmaf(b * b, b, s3);
    s3 = fmaf(c * c, c, s3); s3 = fmaf(d * d, d, s3);
    ma = fmaxf(ma, fmaxf(fmaxf(fabsf(a), fabsf(b)), fmaxf(fabsf(c), fabsf(d))));
  }

  __shared__ double r1[256], r2[256], r3[256];
  __shared__ float  rm[256];
  r1[tid] = (double)s1; r2[tid] = (double)s2; r3[tid] = (double)s3; rm[tid] = ma;
  __syncthreads();
  for (int off = 128; off > 0; off >>= 1) {
    if (tid < off) {
      r1[tid] += r1[tid + off];
      r2[tid] += r2[tid + off];
      r3[tid] += r3[tid + off];
      rm[tid] = fmaxf(rm[tid], rm[tid + off]);
    }
    __syncthreads();
  }

  if (tid == 0) {
    const double n = (double)DIMEN;
    double S1 = r1[0], S2 = r2[0], S3 = r3[0];
    float  mx = rm[0];
    double mean = S1 / n;
    double var  = (S2 - n * mean * mean) / (n - 1.0);       // ddof=1
    if (var < 0.0) var = 0.0;
    float stdv = fmaxf((float)sqrt(var), 1e-9f);
    float nrm  = (float)sqrt(S2 < 0.0 ? 0.0 : S2);
    float norm_val = nrm / (sqrtf((float)DIMEN) + 1e-9f);
    double sd = (double)stdv;
    // E[(x-m)^3] = (Σx³ - 3m Σx² + 2nm³)/n
    double m3 = (S3 - 3.0 * mean * S2 + 2.0 * n * mean * mean * mean) / n;
    float skew = (float)(m3 / (sd * sd * sd));

    float f[6];
    f[0] = tanhf((float)mean);
    f[1] = tanhf(log1pf(stdv));
    f[2] = tanhf(norm_val);
    f[3] = tanhf(log1pf(mx));
    f[4] = tanhf(skew * (1.f / 3.f));
    f[5] = tanhf(log1pf(nrm));
    const float PI = 3.14159265358979323846f;
#pragma unroll
    for (int j = 0; j < 6; ++j) {
      float v = f[j];
      if (v != v) v = 0.f;
      v = (v + 1.f) * 0.5f * PI;
      angles[row * 6 + j] = fminf(fmaxf(v, 0.f), PI);
    }
  }
}

// ---------------------------------------------------------------------------
// Kernel B: build the fixed 64x64 unitary of the StronglyEntanglingLayers.
// Thread t owns basis column t (LDS-resident). CNOTs are index swaps.
// Qubit q <-> bit (5-q) of the flattened statevector index.
// ---------------------------------------------------------------------------
__global__ __launch_bounds__(64) void build_unitary(const float* __restrict__ w,
                                                    float* __restrict__ Ur,
                                                    float* __restrict__ Ui) {
  __shared__ float sr[64 * 65];
  __shared__ float si[64 * 65];
  const int t = threadIdx.x;  // column 0..63
  float* cr = &sr[t * 65];
  float* ci = &si[t * 65];
  for (int b = 0; b < 64; ++b) { cr[b] = (b == t) ? 1.f : 0.f; ci[b] = 0.f; }

  for (int l = 0; l < NLAYERS; ++l) {
    for (int q = 0; q < NQ; ++q) {
      const float* wp = w + (l * NQ + q) * 3;
      float phi = wp[0], th = wp[1], om = wp[2];
      float c  = cosf(0.5f * th),        s  = sinf(0.5f * th);
      float ca = cosf(0.5f * (phi + om)), sa = sinf(0.5f * (phi + om));
      float cb = cosf(0.5f * (phi - om)), sb = sinf(0.5f * (phi - om));
      // Rot = [[e^{-i(phi+om)/2} c, -e^{i(phi-om)/2} s],
      //        [e^{-i(phi-om)/2} s,  e^{i(phi+om)/2} c]]
      float u00r =  ca * c, u00i = -sa * c;
      float u01r = -cb * s, u01i = -sb * s;
      float u10r =  cb * s, u10i = -sb * s;
      float u11r =  ca * c, u11i =  sa * c;
      int mask = 1 << (5 - q);
      for (int b = 0; b < 64; ++b) {
        if (b & mask) continue;
        int b1 = b | mask;
        float a0r = cr[b], a0i = ci[b], a1r = cr[b1], a1i = ci[b1];
        cr[b]  = u00r * a0r - u00i * a0i + u01r * a1r - u01i * a1i;
        ci[b]  = u00r * a0i + u00i * a0r + u01r * a1i + u01i * a1r;
        cr[b1] = u10r * a0r - u10i * a0i + u11r * a1r - u11i * a1i;
        ci[b1] = u10r * a0i + u10i * a0r + u11r * a1i + u11i * a1r;
      }
    }
    int r = (l % (NQ - 1)) + 1;
    for (int i = 0; i < NQ; ++i) {
      int cm = 1 << (5 - i);
      int tm = 1 << (5 - ((i + r) % NQ));
      for (int b = 0; b < 64; ++b) {
        if ((b & cm) && !(b & tm)) {
          int b2 = b | tm;
          float tr = cr[b]; cr[b] = cr[b2]; cr[b2] = tr;
          float ti = ci[b]; ci[b] = ci[b2]; ci[b2] = ti;
        }
      }
    }
  }
  for (int m = 0; m < 64; ++m) {   // U[m][col]
    Ur[m * 64 + t] = cr[m];
    Ui[m * 64 + t] = ci[m];
  }
}

// ---------------------------------------------------------------------------
// Kernel C: S(64 x B) = U (64x64 complex) @ S0 (product states), via
// V_WMMA_F32_16X16X4_F32 (4 real GEMMs), then expectations from LDS.
// Block = 64 threads (2 waves), 16 samples per wave, 32 samples per block.
// ---------------------------------------------------------------------------
#define USTRIDE 72   // column-major U in LDS, padded: bank-conflict-free A reads
#define SSTRIDE 65   // per-sample state rows, padded

__device__ __forceinline__ float out_xf(float v) {
  if (v != v) v = 0.5f;            // nan_to_num(out, nan=0.5)
  v = (v + 1.f) * 0.5f;
  return v * v;
}

__global__ __launch_bounds__(64) void circuit_kernel(const float* __restrict__ angles,
                                                     const float* __restrict__ Ur,
                                                     const float* __restrict__ Ui,
                                                     float* __restrict__ out) {
  __shared__ float sUr[64 * USTRIDE];   // sUr[k*USTRIDE + m] = U[m][k].re
  __shared__ float sUi[64 * USTRIDE];
  __shared__ float sSr[32 * SSTRIDE];   // sSr[sample_local*SSTRIDE + m]
  __shared__ float sSi[32 * SSTRIDE];

  const int tid  = threadIdx.x;
  const int lane = tid & 31;
  const int wave = tid >> 5;
  const int S0   = blockIdx.x * 32;

  // cooperative transposed load of U into LDS
  for (int i = tid; i < 64 * 64; i += 64) {
    int m = i >> 6, k = i & 63;
    sUr[k * USTRIDE + m] = Ur[i];
    sUi[k * USTRIDE + m] = Ui[i];
  }
  __syncthreads();

  // --- B fragments: product state after RX embedding -----------------------
  // B layout (mirror of 32-bit A 16x4): lanes 0-15 hold K = 4kc+{0,1} for
  // column N=lane; lanes 16-31 hold K = 4kc+{2,3} for N=lane-16.
  const int n      = lane & 15;
  const int sample = S0 + wave * 16 + n;
  const int khalf  = (lane >> 4) << 1;   // 0 or 2

  float cq[6], sq[6];
#pragma unroll
  for (int q = 0; q < 6; ++q) {
    float a = angles[sample * 6 + q];
    cq[q] = cosf(0.5f * a);
    sq[q] = sinf(0.5f * a);
  }

  float brx[16], bry[16], bix[16], biy[16];
#pragma unroll
  for (int kc = 0; kc < 16; ++kc) {
#pragma unroll
    for (int kk = 0; kk < 2; ++kk) {
      int k = 4 * kc + khalf + kk;
      float p = 1.f;
#pragma unroll
      for (int q = 0; q < 6; ++q) {
        int bit = (k >> (5 - q)) & 1;
        p *= bit ? sq[q] : cq[q];
      }
      int tm = __popc(k) & 3;            // amp = p * (-i)^popc
      float re = (tm == 0) ? p : ((tm == 2) ? -p : 0.f);
      float im = (tm == 1) ? -p : ((tm == 3) ? p : 0.f);
      if (kk == 0) { brx[kc] = re; bix[kc] = im; }
      else         { bry[kc] = re; biy[kc] = im; }
    }
  }

  // --- complex GEMM: Dr = UrBr - UiBi ; Di = UrBi + UiBr -------------------
  v8f accr[4], acci[4];
  const v8f vz = {0.f, 0.f, 0.f, 0.f, 0.f, 0.f, 0.f, 0.f};
#pragma unroll
  for (int mt = 0; mt < 4; ++mt) { accr[mt] = vz; acci[mt] = vz; }

  const int mrow = lane & 15;
#pragma unroll
  for (int kc = 0; kc < 16; ++kc) {
    v2f br  = { brx[kc],  bry[kc]};
    v2f bi  = { bix[kc],  biy[kc]};
    v2f nbi = {-bix[kc], -biy[kc]};
    const int kb = 4 * kc + khalf;
#pragma unroll
    for (int mt = 0; mt < 4; ++mt) {
      int m = 16 * mt + mrow;
      v2f ar = {sUr[kb * USTRIDE + m], sUr[(kb + 1) * USTRIDE + m]};
      v2f ai = {sUi[kb * USTRIDE + m], sUi[(kb + 1) * USTRIDE + m]};
      accr[mt] = __builtin_amdgcn_wmma_f32_16x16x4_f32(false, ar, false, br,
                                                       (short)0, accr[mt], false, false);
      accr[mt] = __builtin_amdgcn_wmma_f32_16x16x4_f32(false, ai, false, nbi,
                                                       (short)0, accr[mt], false, false);
      acci[mt] = __builtin_amdgcn_wmma_f32_16x16x4_f32(false, ar, false, bi,
                                                       (short)0, acci[mt], false, false);
      acci[mt] = __builtin_amdgcn_wmma_f32_16x16x4_f32(false, ai, false, br,
                                                       (short)0, acci[mt], false, false);
    }
  }

  // --- dump D tiles per-sample into LDS (C/D layout: lane=N, vgpr=M row) ---
  const int nloc  = wave * 16 + n;
  const int mhalf = (lane >> 4) * 8;
#pragma unroll
  for (int mt = 0; mt < 4; ++mt) {
#pragma unroll
    for (int j = 0; j < 8; ++j) {
      sSr[nloc * SSTRIDE + 16 * mt + mhalf + j] = accr[mt][j];
      sSi[nloc * SSTRIDE + 16 * mt + mhalf + j] = acci[mt][j];
    }
  }
  __syncthreads();

  // --- expectations: X=2Re z, Y=2Im z (z = Σ conj(s0)s1), Z = Σ(±)|s|² -----
  if (tid < 32) {
    const float* pr = &sSr[tid * SSTRIDE];
    const float* pi = &sSi[tid * SSTRIDE];
    const int gs = S0 + tid;
#pragma unroll
    for (int q = 0; q < 6; ++q) {
      const int mask = 1 << (5 - q);
      float zr = 0.f, zi = 0.f, zz = 0.f;
      for (int b = 0; b < 64; ++b) {
        float p2 = pr[b] * pr[b] + pi[b] * pi[b];
        zz += (b & mask) ? -p2 : p2;
      }
      for (int b = 0; b < 64; ++b) {
        if (b & mask) continue;
        int b1 = b | mask;
        zr += pr[b] * pr[b1] + pi[b] * pi[b1];
        zi += pr[b] * pi[b1] - pi[b] * pr[b1];
      }
      out[gs * 18 + q]      = out_xf(2.f * zr);
      out[gs * 18 + 6 + q]  = out_xf(2.f * zi);
      out[gs * 18 + 12 + q] = out_xf(zz);
    }
  }
}

// ---------------------------------------------------------------------------
extern "C" void kernel_launch(void* const* d_in, const int* in_sizes, int n_in,
                              void* d_out, int out_size, void* d_ws, size_t ws_size,
                              hipStream_t stream) {
  (void)in_sizes; (void)n_in; (void)out_size; (void)ws_size;
  const float* x = (const float*)d_in[0];   // (1024, 100000) f32
  const float* w = (const float*)d_in[1];   // (5, 6, 3) f32
  float* out = (float*)d_out;               // (1024, 18) f32
  float* ws  = (float*)d_ws;
  float* angles = ws;                       // 1024*6 floats
  float* Ur     = ws + 8192;                // 64*64 floats
  float* Ui     = ws + 12288;               // 64*64 floats

  feat_kernel<<<BATCH, 256, 0, stream>>>(x, angles);
  build_unitary<<<1, 64, 0, stream>>>(w, Ur, Ui);
  circuit_kernel<<<BATCH / 32, 64, 0, stream>>>(angles, Ur, Ui, out);
}